// RMambaMiniFeatureExtractor_44925357916645
// MI455X (gfx1250) — compile-verified
//
#include <hip/hip_runtime.h>
#include <math.h>

typedef __attribute__((ext_vector_type(2))) float v2f;
typedef __attribute__((ext_vector_type(8))) float v8f;

__device__ __forceinline__ float sigmoidf_(float x) { return 1.f / (1.f + expf(-x)); }
__device__ __forceinline__ float siluf_(float x)    { return x / (1.f + expf(-x)); }
__device__ __forceinline__ float softplusf_(float x){ return (x > 20.f) ? x : log1pf(expf(x)); }

// ---------------------------------------------------------------------------
// fp32 GEMM on V_WMMA_F32_16X16X4_F32.  D[M,N] (LDS) = (acc?D:0) + (alphaA*A)*B
// A: row-major [M,K] (LDS or global).  B fetch: BT==0 -> B[k*ldb+n] ; BT==1 -> B[n*ldb+k]
// Fragment layouts per CDNA5 ISA 7.12.2 (wave32):
//   A 16x4 : all lanes m=lane&15 ; VGPR0/1 = K {0,1} (lanes0-15) / {2,3} (lanes16-31)
//   B 4x16 : all lanes n=lane&15 ; K split identically by lane half
//   C/D    : VGPR r -> row r + 8*(lane>>4), col = lane&15
// ---------------------------------------------------------------------------
template <int BT>
__device__ void gemm_w(float* __restrict__ D, int ldd,
                       const float* __restrict__ A, int lda,
                       const float* __restrict__ B, int ldb,
                       int M, int N, int K, float alphaA, bool accin)
{
  const int tid = threadIdx.x;
  const int wid = tid >> 5;
  const int lane = tid & 31;
  const int nwaves = blockDim.x >> 5;
  const int lm = lane & 15;   // A row / B col / C col within tile
  const int kh = lane >> 4;   // K-half (and +8 row offset for C/D)
  const int mt = (M + 15) >> 4, nt = (N + 15) >> 4;
  for (int t = wid; t < mt * nt; t += nwaves) {
    const int tm = (t / nt) * 16, tn = (t % nt) * 16;
    v8f c = {0.f, 0.f, 0.f, 0.f, 0.f, 0.f, 0.f, 0.f};
    if (accin) {
#pragma unroll
      for (int r = 0; r < 8; ++r) {
        int row = tm + r + 8 * kh, col = tn + lm;
        c[r] = (row < M && col < N) ? D[row * ldd + col] : 0.f;
      }
    }
    const int ar = tm + lm;
    const int bc = tn + lm;
    for (int k0 = 0; k0 < K; k0 += 4) {
      const int ak = k0 + 2 * kh;
      v2f a, b;
      a.x = (ar < M && ak     < K) ? alphaA * A[ar * lda + ak]     : 0.f;
      a.y = (ar < M && ak + 1 < K) ? alphaA * A[ar * lda + ak + 1] : 0.f;
      if (BT == 0) {
        b.x = (ak     < K && bc < N) ? B[ak * ldb + bc]       : 0.f;
        b.y = (ak + 1 < K && bc < N) ? B[(ak + 1) * ldb + bc] : 0.f;
      } else {
        b.x = (ak     < K && bc < N) ? B[bc * ldb + ak]     : 0.f;
        b.y = (ak + 1 < K && bc < N) ? B[bc * ldb + ak + 1] : 0.f;
      }
      c = __builtin_amdgcn_wmma_f32_16x16x4_f32(false, a, false, b, (short)0, c,
                                                false, false);
    }
#pragma unroll
    for (int r = 0; r < 8; ++r) {
      int row = tm + r + 8 * kh, col = tn + lm;
      if (row < M && col < N) D[row * ldd + col] = c[r];
    }
  }
}

// ---------------------------------------------------------------------------
// DFT twiddles (ortho), sign of -sin folded in:  Fre = TWC@h, Fim = TWS@h
// ---------------------------------------------------------------------------
__global__ void k_twiddle(float* __restrict__ twc, float* __restrict__ tws)
{
  int gid = blockIdx.x * 256 + threadIdx.x;
  if (gid >= 6400) return;
  int k = gid / 80, l = gid - k * 80;
  int r = (k * l) % 80;
  float ang = 6.28318530717958647692f * (float)r / 80.f;
  float inv = rsqrtf(80.f);
  twc[gid] = cosf(ang) * inv;
  tws[gid] = -sinf(ang) * inv;
}

// ---------------------------------------------------------------------------
// Fused 7x7 s2 conv + BN + ReLU + maxpool(3,2,1).  mode 0: x (3ch); mode 1:
// temporal diffs (12ch) built on the fly.  One WG per image; input halo and
// conv plane live in LDS (136 KB -- CDNA5 320 KB LDS).
// ---------------------------------------------------------------------------
__global__ __launch_bounds__(256)
void k_stem7(const float* __restrict__ x, int mode,
             const float* __restrict__ w, const float* __restrict__ bias,
             const float* __restrict__ g, const float* __restrict__ be,
             const float* __restrict__ m, const float* __restrict__ v,
             float* __restrict__ outp)
{
  extern __shared__ float smem7[];
  float* plane = smem7;           // 4*64*64 = 16384
  float* sbuf  = smem7 + 16384;   // 133*133 = 17689
  const int img = blockIdx.x;
  const int n = img / 160, t = img - n * 160;
  const int tid = threadIdx.x;
  const int nci = mode ? 12 : 3;

  for (int idx = tid; idx < 16384; idx += 256) plane[idx] = bias[idx >> 12];
  __syncthreads();

  for (int ci = 0; ci < nci; ++ci) {
    for (int idx = tid; idx < 17689; idx += 256) {
      int iyy = idx / 133, ixx = idx - iyy * 133;
      int iy = iyy - 3, ix = ixx - 3;
      float val = 0.f;
      if (iy >= 0 && iy < 128 && ix >= 0 && ix < 128) {
        if (mode == 0) {
          val = x[(((size_t)img * 3 + ci) * 128 + iy) * 128 + ix];
        } else {
          int gg = ci / 3, c = ci - gg * 3;
          int tm2 = t - 2 < 0 ? 0 : t - 2, tm1 = t - 1 < 0 ? 0 : t - 1;
          int tp1 = t + 1 > 159 ? 159 : t + 1, tp2 = t + 2 > 159 ? 159 : t + 2;
          int ta, tb2;
          if (gg == 0)      { ta = tm1; tb2 = tm2; }
          else if (gg == 1) { ta = t;   tb2 = tm1; }
          else if (gg == 2) { ta = tp1; tb2 = t;   }
          else              { ta = tp2; tb2 = tp1; }
          val = x[(((size_t)(n * 160 + ta)  * 3 + c) * 128 + iy) * 128 + ix]
              - x[(((size_t)(n * 160 + tb2) * 3 + c) * 128 + iy) * 128 + ix];
        }
      }
      sbuf[idx] = val;
    }
    __syncthreads();
    for (int idx = tid; idx < 16384; idx += 256) {
      int oc = idx >> 12, oy = (idx >> 6) & 63, ox = idx & 63;
      const float* wc = w + ((size_t)oc * nci + ci) * 49;
      float acc = 0.f;
#pragma unroll
      for (int ky = 0; ky < 7; ++ky) {
        const float* srow = sbuf + (oy * 2 + ky) * 133 + ox * 2;
#pragma unroll
        for (int kx = 0; kx < 7; ++kx) acc += wc[ky * 7 + kx] * srow[kx];
      }
      plane[idx] += acc;
    }
    __syncthreads();
  }
  for (int idx = tid; idx < 16384; idx += 256) {
    int oc = idx >> 12;
    float sc = g[oc] * rsqrtf(v[oc] + 1e-5f);
    plane[idx] = fmaxf((plane[idx] - m[oc]) * sc + be[oc], 0.f);
  }
  __syncthreads();
  for (int idx = tid; idx < 4096; idx += 256) {
    int oc = idx >> 10, py = (idx >> 5) & 31, px = idx & 31;
    const float* pl = plane + oc * 4096;
    float mx = -3.4e38f;
#pragma unroll
    for (int ky = 0; ky < 3; ++ky) {
      int iy = py * 2 - 1 + ky; if ((unsigned)iy >= 64u) continue;
#pragma unroll
      for (int kx = 0; kx < 3; ++kx) {
        int ix = px * 2 - 1 + kx; if ((unsigned)ix >= 64u) continue;
        mx = fmaxf(mx, pl[iy * 64 + ix]);
      }
    }
    outp[(size_t)img * 4096 + idx] = mx;
  }
}

// ---------------------------------------------------------------------------
// Fused s21/s22 3x3 convs + BN + ReLU + maxpool + alpha/beta combine.
// ---------------------------------------------------------------------------
__global__ __launch_bounds__(256)
void k_s21s22(const float* __restrict__ xo, const float* __restrict__ xd,
              const float* __restrict__ w21, const float* __restrict__ b21,
              const float* __restrict__ g21, const float* __restrict__ be21,
              const float* __restrict__ m21, const float* __restrict__ v21,
              const float* __restrict__ w22, const float* __restrict__ b22,
              const float* __restrict__ g22, const float* __restrict__ be22,
              const float* __restrict__ m22, const float* __restrict__ v22,
              const float* __restrict__ apha, const float* __restrict__ belta,
              float* __restrict__ outp)
{
  extern __shared__ float smem2[];
  float* sA = smem2;           // 4096 : p1in = a*xo + b*xd
  float* sB = smem2 + 4096;    // 4096 : xd
  float* pA = smem2 + 8192;    // 8192
  float* pB = smem2 + 16384;   // 8192
  const int img = blockIdx.x, tid = threadIdx.x;
  const float ap = apha[0], bl = belta[0];
  for (int idx = tid; idx < 4096; idx += 256) {
    float xov = xo[(size_t)img * 4096 + idx], xdv = xd[(size_t)img * 4096 + idx];
    sA[idx] = ap * xov + bl * xdv;
    sB[idx] = xdv;
  }
  __syncthreads();
  for (int idx = tid; idx < 8192; idx += 256) {
    int oc = idx >> 10, oy = (idx >> 5) & 31, ox = idx & 31;
    float a1 = b21[oc], a2 = b22[oc];
    for (int ci = 0; ci < 4; ++ci) {
#pragma unroll
      for (int ky = 0; ky < 3; ++ky) {
        int iy = oy - 1 + ky; if ((unsigned)iy >= 32u) continue;
#pragma unroll
        for (int kx = 0; kx < 3; ++kx) {
          int ix = ox - 1 + kx; if ((unsigned)ix >= 32u) continue;
          int wi = ((oc * 4 + ci) * 3 + ky) * 3 + kx;
          float sv = sA[ci * 1024 + iy * 32 + ix];
          float dv = sB[ci * 1024 + iy * 32 + ix];
          a1 += w21[wi] * sv;
          a2 += w22[wi] * dv;
        }
      }
    }
    float s1 = g21[oc] * rsqrtf(v21[oc] + 1e-5f);
    float s2 = g22[oc] * rsqrtf(v22[oc] + 1e-5f);
    pA[idx] = fmaxf((a1 - m21[oc]) * s1 + be21[oc], 0.f);
    pB[idx] = fmaxf((a2 - m22[oc]) * s2 + be22[oc], 0.f);
  }
  __syncthreads();
  for (int idx = tid; idx < 2048; idx += 256) {
    int oc = idx >> 8, py = (idx >> 4) & 15, px = idx & 15;
    const float* qa = pA + oc * 1024;
    const float* qb = pB + oc * 1024;
    float mA = -3.4e38f, mB = -3.4e38f;
#pragma unroll
    for (int ky = 0; ky < 3; ++ky) {
      int iy = py * 2 - 1 + ky; if ((unsigned)iy >= 32u) continue;
#pragma unroll
      for (int kx = 0; kx < 3; ++kx) {
        int ix = px * 2 - 1 + kx; if ((unsigned)ix >= 32u) continue;
        mA = fmaxf(mA, qa[iy * 32 + ix]);
        mB = fmaxf(mB, qb[iy * 32 + ix]);
      }
    }
    outp[(size_t)img * 2048 + idx] = ap * mA + bl * mB;
  }
}

// ---------------------------------------------------------------------------
// 3D conv (2,5,5) stride (2,1,1) pad (0,2,2) + bias + BN.  WG per (n, t_out).
// ---------------------------------------------------------------------------
__global__ __launch_bounds__(256)
void k_s3(const float* __restrict__ so, const float* __restrict__ w,
          const float* __restrict__ bias, const float* __restrict__ g,
          const float* __restrict__ be, const float* __restrict__ m,
          const float* __restrict__ v, float* __restrict__ h3)
{
  extern __shared__ float sbuf3[];   // 8ci * 2t * 16 * 16 = 4096
  const int wg = blockIdx.x, n = wg / 80, to = wg - n * 80, tid = threadIdx.x;
  for (int idx = tid; idx < 4096; idx += 256) {
    int ci = idx >> 9, kt = (idx >> 8) & 1, sp = idx & 255;
    sbuf3[idx] = so[((size_t)(n * 160 + 2 * to + kt) * 8 + ci) * 256 + sp];
  }
  __syncthreads();
  for (int idx = tid; idx < 8192; idx += 256) {
    int co = idx >> 8, oy = (idx >> 4) & 15, ox = idx & 15;
    float acc = bias[co];
    for (int ci = 0; ci < 8; ++ci) {
      for (int kt = 0; kt < 2; ++kt) {
        const float* sb = sbuf3 + (ci * 2 + kt) * 256;
        const float* wb = w + (((size_t)co * 8 + ci) * 2 + kt) * 25;
#pragma unroll
        for (int ky = 0; ky < 5; ++ky) {
          int iy = oy - 2 + ky; if ((unsigned)iy >= 16u) continue;
#pragma unroll
          for (int kx = 0; kx < 5; ++kx) {
            int ix = ox - 2 + kx; if ((unsigned)ix >= 16u) continue;
            acc += wb[ky * 5 + kx] * sb[iy * 16 + ix];
          }
        }
      }
    }
    float sc = g[co] * rsqrtf(v[co] + 1e-5f);
    h3[(((size_t)n * 32 + co) * 80 + to) * 256 + (idx & 255)] =
        (acc - m[co]) * sc + be[co];
  }
}

// ---------------------------------------------------------------------------
// Sigmoid-mask spatial pooling -> tokens [B=4][L=80][C=32]
// ---------------------------------------------------------------------------
__global__ __launch_bounds__(256)
void k_mask(const float* __restrict__ h3, float* __restrict__ tok)
{
  int gid = blockIdx.x * 256 + threadIdx.x;
  if (gid >= 10240) return;
  int b = gid / 2560, r = gid - b * 2560, c = r / 80, t = r - c * 80;
  const float* base = h3 + (((size_t)b * 32 + c) * 80 + t) * 256;
  float ssum = 0.f;
  for (int p = 0; p < 256; ++p) ssum += sigmoidf_(base[p]);
  float acc = 0.f;
  for (int p = 0; p < 256; ++p) { float hv = base[p]; acc += hv * sigmoidf_(hv); }
  tok[((size_t)b * 80 + t) * 32 + c] = acc * (128.f / ssum) * (1.f / 256.f);
}

// ---------------------------------------------------------------------------
// Persistent 24-layer Mamba+FFN stack.  4 WGs (one per batch), 8 wave32s,
// all activations in ~195 KB LDS.  GEMMs on V_WMMA_F32_16X16X4_F32.
// ---------------------------------------------------------------------------
struct BlockParams {
  const float *ln1_g, *ln1_b, *ln2_g, *ln2_b;
  const float *in_w, *conv_w, *conv_b, *xproj_w, *dt_w, *dt_b, *A_log, *Dp, *out_w;
  const float *fc1_w, *fc2_w, *r, *i, *rb, *ib;
  const float *bn1_g, *bn1_b, *bn1_m, *bn1_v, *bn2_g, *bn2_b, *bn2_m, *bn2_v;
};

__device__ __forceinline__ void ln80x32(const float* __restrict__ in,
                                        float* __restrict__ o,
                                        const float* __restrict__ g,
                                        const float* __restrict__ bb)
{
  for (int l = threadIdx.x; l < 80; l += blockDim.x) {
    const float* row = in + l * 32;
    float mu = 0.f;
#pragma unroll
    for (int c = 0; c < 32; ++c) mu += row[c];
    mu *= (1.f / 32.f);
    float var = 0.f;
#pragma unroll
    for (int c = 0; c < 32; ++c) { float d = row[c] - mu; var += d * d; }
    float inv = rsqrtf(var * (1.f / 32.f) + 1e-5f);
    float* orow = o + l * 32;
#pragma unroll
    for (int c = 0; c < 32; ++c) orow[c] = (row[c] - mu) * inv * g[c] + bb[c];
  }
}

__global__ __launch_bounds__(256)
void k_blocks(const float* __restrict__ tok, const float* __restrict__ TWC,
              const float* __restrict__ TWS, BlockParams P,
              float* __restrict__ out)
{
  extern __shared__ float sm[];
  float* Xs  = sm;          // 80x32 tokens (residual stream)
  float* XN  = sm + 2560;   // 80x32 layernorm / proj out
  float* XM  = sm + 5120;   // 80x64 xm ; later g (iDFT out)
  float* Zs  = sm + 10240;  // 80x64 z
  float* XC  = sm + 15360;  // 80x64 conv+silu
  float* DTb = sm + 20480;  // 80x64 dt ; later re2
  float* Yb  = sm + 25600;  // 80x64 y ; later h1
  float* FR  = sm + 30720;  // 80x64 DFT real
  float* FI  = sm + 35840;  // 80x64 DFT imag
  float* G2  = sm + 40960;  // 80x64 im2
  float* DB  = sm + 46080;  // 80x34 dt|B|C
  const int tid = threadIdx.x;
  const int b = blockIdx.x;

  for (int i2 = tid; i2 < 2560; i2 += 256) Xs[i2] = tok[b * 2560 + i2];
  __syncthreads();

  for (int L = 0; L < 24; ++L) {
    const float* ln1g = P.ln1_g + L * 32; const float* ln1b = P.ln1_b + L * 32;
    const float* ln2g = P.ln2_g + L * 32; const float* ln2b = P.ln2_b + L * 32;
    const float* inw  = P.in_w  + L * 4096;
    const float* cw   = P.conv_w + L * 256; const float* cb = P.conv_b + L * 64;
    const float* xpw  = P.xproj_w + L * 2176;
    const float* dtw  = P.dt_w + L * 128;  const float* dtb = P.dt_b + L * 64;
    const float* alog = P.A_log + L * 1024; const float* dpp = P.Dp + L * 64;
    const float* outw = P.out_w + L * 2048;
    const float* f1w  = P.fc1_w + L * 2048; const float* f2w = P.fc2_w + L * 2048;
    const float* rw   = P.r + L * 4096;  const float* iw = P.i + L * 4096;
    const float* rbp  = P.rb + L * 64;   const float* ibp = P.ib + L * 64;
    const float* b1g  = P.bn1_g + L * 64; const float* b1b = P.bn1_b + L * 64;
    const float* b1m  = P.bn1_m + L * 64; const float* b1v = P.bn1_v + L * 64;
    const float* b2g  = P.bn2_g + L * 32; const float* b2b = P.bn2_b + L * 32;
    const float* b2m  = P.bn2_m + L * 32; const float* b2v = P.bn2_v + L * 32;

    if (L < 23) {  // warm L2 for next layer's big weights
      __builtin_prefetch(P.r + (L + 1) * 4096, 0, 1);
      __builtin_prefetch(P.i + (L + 1) * 4096, 0, 1);
    }

    // ---------------- Mamba ----------------
    ln80x32(Xs, XN, ln1g, ln1b);
    __syncthreads();
    // xz = XN @ in_w^T  (split into xm / z)
    gemm_w<1>(XM, 64, XN, 32, inw,           32, 80, 64, 32, 1.f, false);
    gemm_w<1>(Zs, 64, XN, 32, inw + 64 * 32, 32, 80, 64, 32, 1.f, false);
    __syncthreads();
    // causal depthwise conv1d (k=4) + silu
    for (int idx = tid; idx < 5120; idx += 256) {
      int l = idx >> 6, e = idx & 63;
      float a = cb[e];
#pragma unroll
      for (int j = 0; j < 4; ++j) {
        int ls = l - 3 + j;
        if (ls >= 0) a += cw[e * 4 + j] * XM[ls * 64 + e];
      }
      XC[idx] = siluf_(a);
    }
    __syncthreads();
    // dbl = xc @ xproj^T  [80,34]
    gemm_w<1>(DB, 34, XC, 64, xpw, 64, 80, 34, 64, 1.f, false);
    __syncthreads();
    // dt = softplus(dbl[:, :2] @ dt_w^T + dt_b)
    for (int idx = tid; idx < 5120; idx += 256) {
      int l = idx >> 6, e = idx & 63;
      float t0 = DB[l * 34] * dtw[e * 2] + DB[l * 34 + 1] * dtw[e * 2 + 1] + dtb[e];
      DTb[idx] = softplusf_(t0);
    }
    __syncthreads();
    // selective scan: 64 chains, states in registers, fused C-projection
    if (tid < 64) {
      const int d = tid;
      float negA[16], h[16];
#pragma unroll
      for (int n = 0; n < 16; ++n) { negA[n] = -expf(alog[d * 16 + n]); h[n] = 0.f; }
      const float Dv = dpp[d];
      for (int l = 0; l < 80; ++l) {
        float dtv = DTb[l * 64 + d], xcv = XC[l * 64 + d], yv = 0.f;
        const float* db = DB + l * 34;
#pragma unroll
        for (int n = 0; n < 16; ++n) {
          float dA = expf(dtv * negA[n]);
          h[n] = dA * h[n] + dtv * db[2 + n] * xcv;
          yv += h[n] * db[18 + n];
        }
        Yb[l * 64 + d] = yv + Dv * xcv;
      }
    }
    __syncthreads();
    for (int idx = tid; idx < 5120; idx += 256) Yb[idx] *= siluf_(Zs[idx]);
    __syncthreads();
    gemm_w<1>(XN, 32, Yb, 64, outw, 64, 80, 32, 64, 1.f, false);
    __syncthreads();
    for (int idx = tid; idx < 2560; idx += 256) Xs[idx] += XN[idx];
    __syncthreads();

    // ---------------- Frequency-domain FFN ----------------
    ln80x32(Xs, XN, ln2g, ln2b);
    __syncthreads();
    gemm_w<1>(Yb, 64, XN, 32, f1w, 32, 80, 64, 32, 1.f, false);   // h1
    __syncthreads();
    for (int idx = tid; idx < 5120; idx += 256) {
      int n = idx & 63;
      float v1 = (Yb[idx] - b1m[n]) * b1g[n] * rsqrtf(b1v[n] + 1e-5f) + b1b[n];
      Yb[idx] = fmaxf(v1, 0.f);
    }
    __syncthreads();
    // DFT (ortho): Fre = TWC@h1, Fim = TWS@h1  (A matrices in global/L2)
    gemm_w<0>(FR, 64, TWC, 80, Yb, 64, 80, 64, 80, 1.f, false);
    gemm_w<0>(FI, 64, TWS, 80, Yb, 64, 80, 64, 80, 1.f, false);
    __syncthreads();
    // re2 = Fre@r - Fim@i ; im2 = Fim@r + Fre@i  (minus folded into A load)
    gemm_w<0>(DTb, 64, FR, 64, rw, 64, 80, 64, 64,  1.f, false);
    __syncthreads();
    gemm_w<0>(DTb, 64, FI, 64, iw, 64, 80, 64, 64, -1.f, true);
    __syncthreads();
    gemm_w<0>(G2,  64, FI, 64, rw, 64, 80, 64, 64,  1.f, false);
    __syncthreads();
    gemm_w<0>(G2,  64, FR, 64, iw, 64, 80, 64, 64,  1.f, true);
    __syncthreads();
    for (int idx = tid; idx < 5120; idx += 256) {
      int n = idx & 63;
      DTb[idx] = fmaxf(DTb[idx] + rbp[n], 0.f);
      G2[idx]  = fmaxf(G2[idx]  + ibp[n], 0.f);
    }
    __syncthreads();
    // g = real(iDFT) = TWC@re2 + TWS@im2   (cos/sin symmetric)
    gemm_w<0>(XM, 64, TWC, 80, DTb, 64, 80, 64, 80, 1.f, false);
    __syncthreads();
    gemm_w<0>(XM, 64, TWS, 80, G2,  64, 80, 64, 80, 1.f, true);
    __syncthreads();
    gemm_w<1>(XN, 32, XM, 64, f2w, 64, 80, 32, 64, 1.f, false);
    __syncthreads();
    for (int idx = tid; idx < 2560; idx += 256) {
      int c = idx & 31;
      float v1 = (XN[idx] - b2m[c]) * b2g[c] * rsqrtf(b2v[c] + 1e-5f) + b2b[c];
      Xs[idx] += fmaxf(v1, 0.f);
    }
    __syncthreads();
  }

  // out [4,32,160]: transpose + nearest upsample x2 along time
  for (int i2 = tid; i2 < 2560; i2 += 256) {
    int l = i2 >> 5, c = i2 & 31;
    float v1 = Xs[i2];
    out[((size_t)b * 32 + c) * 160 + 2 * l]     = v1;
    out[((size_t)b * 32 + c) * 160 + 2 * l + 1] = v1;
  }
}

// ---------------------------------------------------------------------------
extern "C" void kernel_launch(void* const* d_in, const int* in_sizes, int n_in,
                              void* d_out, int out_size, void* d_ws, size_t ws_size,
                              hipStream_t stream)
{
  (void)in_sizes; (void)n_in; (void)out_size; (void)ws_size;
  const float* x = (const float*)d_in[0];
  // stem params: for each of s11,s12,s21,s22 -> w,b,g,be,m,v
  const float* s11[6]; const float* s12[6]; const float* s21[6]; const float* s22[6];
  for (int j = 0; j < 6; ++j) {
    s11[j] = (const float*)d_in[1 + j];
    s12[j] = (const float*)d_in[7 + j];
    s21[j] = (const float*)d_in[13 + j];
    s22[j] = (const float*)d_in[19 + j];
  }
  const float* s3_w  = (const float*)d_in[25];
  const float* s3_b  = (const float*)d_in[26];
  const float* s3_g  = (const float*)d_in[27];
  const float* s3_be = (const float*)d_in[28];
  const float* s3_m  = (const float*)d_in[29];
  const float* s3_v  = (const float*)d_in[30];
  const float* apha  = (const float*)d_in[31];
  const float* belta = (const float*)d_in[32];

  BlockParams bp;
  bp.ln1_g  = (const float*)d_in[33]; bp.ln1_b = (const float*)d_in[34];
  bp.ln2_g  = (const float*)d_in[35]; bp.ln2_b = (const float*)d_in[36];
  bp.in_w   = (const float*)d_in[37];
  bp.conv_w = (const float*)d_in[38]; bp.conv_b = (const float*)d_in[39];
  bp.xproj_w= (const float*)d_in[40];
  bp.dt_w   = (const float*)d_in[41]; bp.dt_b  = (const float*)d_in[42];
  bp.A_log  = (const float*)d_in[43]; bp.Dp    = (const float*)d_in[44];
  bp.out_w  = (const float*)d_in[45];
  bp.fc1_w  = (const float*)d_in[46]; bp.fc2_w = (const float*)d_in[47];
  bp.r      = (const float*)d_in[48]; bp.i     = (const float*)d_in[49];
  bp.rb     = (const float*)d_in[50]; bp.ib    = (const float*)d_in[51];
  bp.bn1_g  = (const float*)d_in[52]; bp.bn1_b = (const float*)d_in[53];
  bp.bn1_m  = (const float*)d_in[54]; bp.bn1_v = (const float*)d_in[55];
  bp.bn2_g  = (const float*)d_in[56]; bp.bn2_b = (const float*)d_in[57];
  bp.bn2_m  = (const float*)d_in[58]; bp.bn2_v = (const float*)d_in[59];

  float* ws  = (float*)d_ws;
  float* TWC = ws;                  // 6400
  float* TWS = ws + 6400;           // 6400
  float* TOK = ws + 12800;          // 10240
  float* XO  = ws + 23040;          // 640*4*32*32 = 2621440
  float* XD  = XO + 2621440;        // 2621440
  float* SO  = XD + 2621440;        // 640*8*16*16 = 1310720
  float* H3  = SO + 1310720;        // 4*32*80*256 = 2621440

  k_twiddle<<<25, 256, 0, stream>>>(TWC, TWS);
  k_stem7<<<640, 256, (16384 + 17689) * sizeof(float), stream>>>(
      x, 0, s11[0], s11[1], s11[2], s11[3], s11[4], s11[5], XO);
  k_stem7<<<640, 256, (16384 + 17689) * sizeof(float), stream>>>(
      x, 1, s12[0], s12[1], s12[2], s12[3], s12[4], s12[5], XD);
  k_s21s22<<<640, 256, 24576 * sizeof(float), stream>>>(
      XO, XD,
      s21[0], s21[1], s21[2], s21[3], s21[4], s21[5],
      s22[0], s22[1], s22[2], s22[3], s22[4], s22[5],
      apha, belta, SO);
  k_s3<<<320, 256, 4096 * sizeof(float), stream>>>(
      SO, s3_w, s3_b, s3_g, s3_be, s3_m, s3_v, H3);
  k_mask<<<40, 256, 0, stream>>>(H3, TOK);
  k_blocks<<<4, 256, 48800 * sizeof(float), stream>>>(TOK, TWC, TWS, bp, (float*)d_out);
}